// MultiScaleSheafDecomposition_71640054497901
// MI455X (gfx1250) — compile-verified
//
#include <hip/hip_runtime.h>
#include <math.h>

typedef __attribute__((ext_vector_type(2))) float v2f;
typedef __attribute__((ext_vector_type(4))) float v4f;
typedef __attribute__((ext_vector_type(8))) float v8f;

#define EPS 1e-8f
#define LN_EPS 1e-5f

// ---------------- WMMA helper: D = A(16x4,f32) * B(4x16,f32) + C ----------------
static __device__ __forceinline__ v8f wmma4(v2f a, v2f b, v8f c) {
    return __builtin_amdgcn_wmma_f32_16x16x4_f32(false, a, false, b, (short)0, c,
                                                 false, false);
}

static __device__ __forceinline__ float wave_sum(float x) {
    #pragma unroll
    for (int o = 16; o >= 1; o >>= 1) x += __shfl_xor(x, o, 32);
    return x;
}

// ---------------- small utility kernels ----------------
__global__ void zero_kernel(float* p, int n) {
    int i = blockIdx.x * blockDim.x + threadIdx.x;
    if (i < n) p[i] = 0.f;
}

__global__ void deg_kernel(const int* __restrict__ ei, float* __restrict__ deg, int E) {
    int i = blockIdx.x * blockDim.x + threadIdx.x;
    if (i < E) {
        atomicAdd(&deg[ei[i]], 1.f);       // row
        atomicAdd(&deg[ei[E + i]], 1.f);   // col
    }
}

__global__ void init_v_kernel(float* v, int n) {
    int i = blockIdx.x * blockDim.x + threadIdx.x;
    if (i < n) {
        unsigned u = (unsigned)i * 2654435761u ^ 0x9e3779b9u;
        u ^= u >> 16; u *= 0x85ebca6bu; u ^= u >> 13; u *= 0xc2b2ae35u; u ^= u >> 16;
        v[i] = (float)(u & 0xFFFFFFu) * (1.f / 8388608.f) - 1.f;  // [-1,1)
    }
}

// out[i] = deg[i/32] * x[i]
__global__ void degx_kernel(float* __restrict__ out, const float* __restrict__ x,
                            const float* __restrict__ deg, int n) {
    int i = blockIdx.x * blockDim.x + threadIdx.x;
    if (i < n) out[i] = deg[i >> 5] * x[i];
}

// out[row] -= Q[e] @ x[col]  (one wave per edge, lane = output row).
// row/col are wave-uniform -> readfirstlane puts them in SGPRs so the x[col]
// broadcast becomes scalar (K$) loads; Q is streamed with non-temporal B128
// loads so it never pollutes L2 (x, weights and T_k stay resident).
__global__ void edge_matvec_kernel(const float* __restrict__ Q, const float* __restrict__ x,
                                   const int* __restrict__ ei, float* __restrict__ out, int E) {
    int lane = threadIdx.x & 31;
    int e = blockIdx.x * 8 + (threadIdx.x >> 5);
    if (e >= E) return;
    int row = __builtin_amdgcn_readfirstlane(ei[e]);
    int col = __builtin_amdgcn_readfirstlane(ei[E + e]);
    const float* __restrict__ xv = x + (size_t)col * 32;   // uniform address
    const v4f* Q4 = (const v4f*)(Q + (size_t)e * 1024);
    float y = 0.f;
    #pragma unroll
    for (int j4 = 0; j4 < 8; ++j4) {
        v4f q = __builtin_nontemporal_load(&Q4[lane * 8 + j4]);  // stream, don't cache
        y += q.x * xv[4 * j4 + 0];
        y += q.y * xv[4 * j4 + 1];
        y += q.z * xv[4 * j4 + 2];
        y += q.w * xv[4 * j4 + 3];
    }
    atomicAdd(&out[(size_t)row * 32 + lane], -y);
}

// s[0]+=a.b  s[1]+=a.a  s[2]+=b.b
__global__ void dot3_kernel(const float* __restrict__ a, const float* __restrict__ b,
                            float* __restrict__ s, int n) {
    int lane = threadIdx.x & 31;
    float x = 0.f, y = 0.f, z = 0.f;
    for (int i = blockIdx.x * blockDim.x + threadIdx.x; i < n; i += gridDim.x * blockDim.x) {
        float av = a[i], bv = b[i];
        x += av * bv; y += av * av; z += bv * bv;
    }
    x = wave_sum(x); y = wave_sum(y); z = wave_sum(z);
    if (lane == 0) {
        atomicAdd(&s[0], x);
        atomicAdd(&s[1], y);
        atomicAdd(&s[2], z);
    }
}

// s[3] = lambda, s[4] = 1/(||Lv||+eps)
__global__ void pi_finalize_kernel(float* s) {
    s[3] = s[0] / (s[1] + EPS);
    s[4] = 1.f / (sqrtf(s[2]) + EPS);
}

// s[5] = alpha = 2/(max(lambda,1)+eps)
__global__ void alpha_finalize_kernel(float* s) {
    float lam = fmaxf(s[3], 1.f);
    s[5] = 2.f / (lam + EPS);
}

// dst = src * s[4]
__global__ void scale_kernel(float* __restrict__ dst, const float* __restrict__ src,
                             const float* __restrict__ s, int n) {
    int i = blockIdx.x * blockDim.x + threadIdx.x;
    if (i < n) dst[i] = src[i] * s[4];
}

// T1 = alpha*raw - h
__global__ void cheb_t1_kernel(float* __restrict__ T1, const float* __restrict__ raw,
                               const float* __restrict__ h, const float* __restrict__ s, int n) {
    int i = blockIdx.x * blockDim.x + threadIdx.x;
    if (i < n) T1[i] = s[5] * raw[i] - h[i];
}

// Tn = 2*(alpha*raw - Tc) - Tp
__global__ void cheb_next_kernel(float* __restrict__ Tn, const float* __restrict__ raw,
                                 const float* __restrict__ Tc, const float* __restrict__ Tp,
                                 const float* __restrict__ s, int n) {
    int i = blockIdx.x * blockDim.x + threadIdx.x;
    if (i < n) Tn[i] = 2.f * (s[5] * raw[i] - Tc[i]) - Tp[i];
}

__global__ void softmax_kernel(const float* __restrict__ c, float* __restrict__ w) {
    int b = threadIdx.x;
    if (b >= 3) return;
    float m = c[b * 5];
    for (int k = 1; k < 5; ++k) m = fmaxf(m, c[b * 5 + k]);
    float e[5], sm = 0.f;
    for (int k = 0; k < 5; ++k) { e[k] = __expf(c[b * 5 + k] - m); sm += e[k]; }
    for (int k = 0; k < 5; ++k) w[b * 5 + k] = e[k] / sm;
}

// ---------------- per-band projection: p_b = silu(LN( (sum_k w_bk T_k) @ W_b + b_b )) ----------------
// one wave per 16-row tile; V_WMMA_F32_16X16X4_F32, K=32 in 8 steps; band-combine fused into A load.
__global__ void proj_kernel(const float* __restrict__ T0, const float* __restrict__ T1,
                            const float* __restrict__ T2, const float* __restrict__ T3,
                            const float* __restrict__ T4, const float* __restrict__ w,
                            const float* __restrict__ pW, const float* __restrict__ pb,
                            const float* __restrict__ pg, const float* __restrict__ pbeta,
                            float* __restrict__ p, int N) {
    __shared__ float lds[4][16 * 32];
    int lane = threadIdx.x & 31;
    int wave = threadIdx.x >> 5;
    int half = lane >> 4;
    int lr = lane & 15;
    int ntiles = N >> 4;
    int tile = blockIdx.x * 4 + wave;
    bool act = tile < ntiles;
    if (tile >= ntiles) tile = ntiles - 1;
    int row0 = tile * 16;
    int band = blockIdx.y;

    float w0 = w[band * 5 + 0], w1 = w[band * 5 + 1], w2 = w[band * 5 + 2];
    float w3 = w[band * 5 + 3], w4 = w[band * 5 + 4];
    const float* W = pW + band * 1024;

    v8f acc0 = {}, acc1 = {};
    #pragma unroll
    for (int kk = 0; kk < 8; ++kk) {
        int k0 = kk * 4 + (half ? 2 : 0);
        size_t ra = (size_t)(row0 + lr) * 32;
        v2f a;
        a.x = w0 * T0[ra + k0] + w1 * T1[ra + k0] + w2 * T2[ra + k0]
            + w3 * T3[ra + k0] + w4 * T4[ra + k0];
        a.y = w0 * T0[ra + k0 + 1] + w1 * T1[ra + k0 + 1] + w2 * T2[ra + k0 + 1]
            + w3 * T3[ra + k0 + 1] + w4 * T4[ra + k0 + 1];
        v2f b0, b1;
        b0.x = W[k0 * 32 + lr];            b0.y = W[(k0 + 1) * 32 + lr];
        b1.x = W[k0 * 32 + 16 + lr];       b1.y = W[(k0 + 1) * 32 + 16 + lr];
        acc0 = wmma4(a, b0, acc0);
        acc1 = wmma4(a, b1, acc1);
    }

    // spill C tile (+bias) to LDS using the documented C layout
    float bias0 = pb[band * 32 + lr], bias1 = pb[band * 32 + 16 + lr];
    float* t = lds[wave];
    #pragma unroll
    for (int r = 0; r < 8; ++r) {
        int m = r + half * 8;
        t[m * 32 + lr]      = acc0[r] + bias0;
        t[m * 32 + 16 + lr] = acc1[r] + bias1;
    }
    __syncthreads();

    // LayerNorm over D=32 (one row per iteration, 32 lanes = 32 cols) + SiLU
    float g = pg[band * 32 + lane], be = pbeta[band * 32 + lane];
    for (int r = 0; r < 16; ++r) {
        float x = t[r * 32 + lane];
        float sm = wave_sum(x);
        float sq = wave_sum(x * x);
        float mu = sm * (1.f / 32.f);
        float var = sq * (1.f / 32.f) - mu * mu;
        float yv = (x - mu) * rsqrtf(var + LN_EPS) * g + be;
        float sl = yv / (1.f + __expf(-yv));
        if (act) p[(size_t)band * N * 32 + (size_t)(row0 + r) * 32 + lane] = sl;
    }
}

// ---------------- fusion: out = h + silu(LN(cat @ W1 + b1)) @ W2 + b2 ----------------
__global__ void fusion_kernel(const float* __restrict__ p, const float* __restrict__ W1,
                              const float* __restrict__ b1, const float* __restrict__ g1,
                              const float* __restrict__ be1, const float* __restrict__ W2,
                              const float* __restrict__ b2, const float* __restrict__ h,
                              float* __restrict__ out, int N) {
    __shared__ float lds[4][16 * 32];
    int lane = threadIdx.x & 31;
    int wave = threadIdx.x >> 5;
    int half = lane >> 4;
    int lr = lane & 15;
    int ntiles = N >> 4;
    int tile = blockIdx.x * 4 + wave;
    bool act = tile < ntiles;
    if (tile >= ntiles) tile = ntiles - 1;
    int row0 = tile * 16;

    // GEMM1: [16,96] @ [96,32], K in 24 steps of 4; cat gathered from the 3 band buffers
    v8f acc0 = {}, acc1 = {};
    #pragma unroll
    for (int kk = 0; kk < 24; ++kk) {
        int c0 = kk * 4 + (half ? 2 : 0);
        int c1 = c0 + 1;
        size_t r = (size_t)(row0 + lr);
        v2f a;
        a.x = p[(size_t)(c0 >> 5) * N * 32 + r * 32 + (c0 & 31)];
        a.y = p[(size_t)(c1 >> 5) * N * 32 + r * 32 + (c1 & 31)];
        v2f bb0, bb1;
        bb0.x = W1[c0 * 32 + lr];       bb0.y = W1[c1 * 32 + lr];
        bb1.x = W1[c0 * 32 + 16 + lr];  bb1.y = W1[c1 * 32 + 16 + lr];
        acc0 = wmma4(a, bb0, acc0);
        acc1 = wmma4(a, bb1, acc1);
    }

    float bias0 = b1[lr], bias1 = b1[16 + lr];
    float* t = lds[wave];
    #pragma unroll
    for (int r = 0; r < 8; ++r) {
        int m = r + half * 8;
        t[m * 32 + lr]      = acc0[r] + bias0;
        t[m * 32 + 16 + lr] = acc1[r] + bias1;
    }
    __syncthreads();

    // LN + SiLU in place
    float g = g1[lane], be = be1[lane];
    for (int r = 0; r < 16; ++r) {
        float x = t[r * 32 + lane];
        float sm = wave_sum(x);
        float sq = wave_sum(x * x);
        float mu = sm * (1.f / 32.f);
        float var = sq * (1.f / 32.f) - mu * mu;
        float yv = (x - mu) * rsqrtf(var + LN_EPS) * g + be;
        t[r * 32 + lane] = yv / (1.f + __expf(-yv));
    }
    __syncthreads();

    // GEMM2: [16,32] @ [32,32], A from LDS
    v8f d0 = {}, d1 = {};
    #pragma unroll
    for (int kk = 0; kk < 8; ++kk) {
        int k0 = kk * 4 + (half ? 2 : 0);
        v2f a;
        a.x = t[lr * 32 + k0];
        a.y = t[lr * 32 + k0 + 1];
        v2f bb0, bb1;
        bb0.x = W2[k0 * 32 + lr];       bb0.y = W2[(k0 + 1) * 32 + lr];
        bb1.x = W2[k0 * 32 + 16 + lr];  bb1.y = W2[(k0 + 1) * 32 + 16 + lr];
        d0 = wmma4(a, bb0, d0);
        d1 = wmma4(a, bb1, d1);
    }

    float bs0 = b2[lr], bs1 = b2[16 + lr];
    if (act) {
        #pragma unroll
        for (int r = 0; r < 8; ++r) {
            int m = r + half * 8;
            size_t idx = (size_t)(row0 + m) * 32;
            out[idx + lr]      = d0[r] + bs0 + h[idx + lr];
            out[idx + 16 + lr] = d1[r] + bs1 + h[idx + 16 + lr];
        }
    }
}

// ---------------- launch ----------------
extern "C" void kernel_launch(void* const* d_in, const int* in_sizes, int n_in,
                              void* d_out, int out_size, void* d_ws, size_t ws_size,
                              hipStream_t stream) {
    const float* h      = (const float*)d_in[0];
    const float* Q      = (const float*)d_in[1];
    const int*   ei     = (const int*)d_in[2];
    const float* coeffs = (const float*)d_in[3];
    const float* pW     = (const float*)d_in[4];
    const float* pb     = (const float*)d_in[5];
    const float* pg     = (const float*)d_in[6];
    const float* pbeta  = (const float*)d_in[7];
    const float* fW1    = (const float*)d_in[8];
    const float* fb1    = (const float*)d_in[9];
    const float* fg     = (const float*)d_in[10];
    const float* fbe    = (const float*)d_in[11];
    const float* fW2    = (const float*)d_in[12];
    const float* fb2    = (const float*)d_in[13];
    float* out = (float*)d_out;
    float* ws  = (float*)d_ws;

    const int D = 32;
    const int N = in_sizes[0] / D;
    const int E = in_sizes[2] / 2;
    const size_t ND = (size_t)N * D;
    const int nd = (int)ND;

    float* deg  = ws;               // N
    float* scal = ws + N;           // 8 scalars: num,den,nrm,lam,inv,alpha
    float* w    = ws + N + 8;       // 15 softmax weights (padded to 24)
    float* v    = ws + N + 32;
    float* Lv   = v  + ND;
    float* raw  = Lv + ND;
    float* T1   = raw + ND;
    float* T2   = T1 + ND;
    float* T3   = T2 + ND;
    float* T4   = T3 + ND;
    float* p    = T4 + ND;          // 3 * ND band outputs

    const int TPB = 256;
    int gN  = (N + TPB - 1) / TPB;
    int gND = (nd + TPB - 1) / TPB;
    int gE  = (E + TPB - 1) / TPB;
    int gEW = (E + 7) / 8;          // one wave per edge, 8 waves/block

    auto matvec = [&](float* dst, const float* x) {
        degx_kernel<<<gND, TPB, 0, stream>>>(dst, x, deg, nd);
        edge_matvec_kernel<<<gEW, TPB, 0, stream>>>(Q, x, ei, dst, E);
    };

    // degrees
    zero_kernel<<<gN, TPB, 0, stream>>>(deg, N);
    zero_kernel<<<1, 32, 0, stream>>>(scal, 8);
    deg_kernel<<<gE, TPB, 0, stream>>>(ei, deg, E);

    // power iteration init: v = hash-normal-ish, normalized
    init_v_kernel<<<gND, TPB, 0, stream>>>(v, nd);
    dot3_kernel<<<512, TPB, 0, stream>>>(v, v, scal, nd);
    pi_finalize_kernel<<<1, 1, 0, stream>>>(scal);
    scale_kernel<<<gND, TPB, 0, stream>>>(v, v, scal, nd);

    // 5 power iterations for lambda_max
    for (int it = 0; it < 5; ++it) {
        matvec(Lv, v);
        zero_kernel<<<1, 32, 0, stream>>>(scal, 3);
        dot3_kernel<<<512, TPB, 0, stream>>>(v, Lv, scal, nd);
        pi_finalize_kernel<<<1, 1, 0, stream>>>(scal);
        scale_kernel<<<gND, TPB, 0, stream>>>(v, Lv, scal, nd);
    }
    alpha_finalize_kernel<<<1, 1, 0, stream>>>(scal);

    // Chebyshev basis T0=h, T1..T4 (shared across bands)
    matvec(raw, h);
    cheb_t1_kernel<<<gND, TPB, 0, stream>>>(T1, raw, h, scal, nd);
    matvec(raw, T1);
    cheb_next_kernel<<<gND, TPB, 0, stream>>>(T2, raw, T1, h, scal, nd);
    matvec(raw, T2);
    cheb_next_kernel<<<gND, TPB, 0, stream>>>(T3, raw, T2, T1, scal, nd);
    matvec(raw, T3);
    cheb_next_kernel<<<gND, TPB, 0, stream>>>(T4, raw, T3, T2, scal, nd);

    // band weights
    softmax_kernel<<<1, 4, 0, stream>>>(coeffs, w);

    // per-band projection (WMMA) — combine + GEMM + LN + SiLU fused
    int ntiles = N / 16;
    dim3 pgrid((ntiles + 3) / 4, 3);
    proj_kernel<<<pgrid, 128, 0, stream>>>(h, T1, T2, T3, T4, w, pW, pb, pg, pbeta, p, N);

    // fusion (WMMA x2) + residual
    fusion_kernel<<<(ntiles + 3) / 4, 128, 0, stream>>>(p, fW1, fb1, fg, fbe, fW2, fb2,
                                                        h, out, N);
}